// GlobalSAG_38817914421917
// MI455X (gfx1250) — compile-verified
//
#include <hip/hip_runtime.h>
#include <hip/hip_bf16.h>

// ---------------------------------------------------------------------------
// GIN (2 layers) + SAGPool + head for MI455X (gfx1250, wave32).
//  - fp32 WMMA (v_wmma_f32_16x16x4_f32) for all GEMMs (top-k selection is
//    value-order sensitive -> keep reference fp32 numerics).
//  - GEMMs templated on compile-time N/K: fully unrolled k-loop, immediate
//    global offsets, no divergence (EXEC all-1s for WMMA).
//  - A tile staged in LDS, shared by the 8 waves of a block (8x reuse);
//    async global->LDS (GLOBAL_LOAD_ASYNC_TO_LDS_B128 + s_wait_asynccnt)
//    where the staging is a pure copy.
//  - Edge scatter-adds via hardware f32 global atomics (all L2-resident:
//    192 MB L2 >> ~80 MB working set) + global_prefetch_b8 lookahead.
// ---------------------------------------------------------------------------

#define HD       128      // hidden dim
#define H2       256      // 2*H
#define NNODES   512      // nodes per graph
#define NGRAPH   128      // graphs
#define NT_TOTAL (NNODES * NGRAPH)
#define KPOOL    256      // ceil(0.5 * N)
#define NOUT     10
#define BN_EPS   1e-5f

typedef __attribute__((ext_vector_type(2))) float v2f;
typedef __attribute__((ext_vector_type(8))) float v8f;
typedef int v4i_vs __attribute__((vector_size(16)));   // matches builtin proto

#if defined(__has_builtin)
#if __has_builtin(__builtin_amdgcn_global_load_async_to_lds_b128) && \
    __has_builtin(__builtin_amdgcn_s_wait_asynccnt)
#define USE_ASYNC_LDS 1
#endif
#endif

// 16-byte global -> LDS copy; async DMA path on gfx1250 when available.
__device__ __forceinline__ void copy16_g2l(const float* __restrict__ gsrc,
                                           float* __restrict__ ldst)
{
#ifdef USE_ASYNC_LDS
    __builtin_amdgcn_global_load_async_to_lds_b128(
        (__attribute__((address_space(1))) v4i_vs*)gsrc,
        (__attribute__((address_space(3))) v4i_vs*)ldst, 0, 0);
#else
    *(float4*)ldst = *(const float4*)gsrc;
#endif
}

__device__ __forceinline__ void wait_async_lds()
{
#ifdef USE_ASYNC_LDS
    __builtin_amdgcn_s_wait_asynccnt(0);
#endif
}

// ---------------------------------------------------------------------------
// Edge scatter-add: agg[dst] += x[src].  One wave per edge, float4 per lane.
// ---------------------------------------------------------------------------
__global__ __launch_bounds__(256)
void gin_scatter_add(const float* __restrict__ x,
                     const int* __restrict__ src,
                     const int* __restrict__ dst,
                     float* __restrict__ agg,
                     int nEdges)
{
    const int lane = threadIdx.x & 31;
    const int wave = threadIdx.x >> 5;
    const int e = blockIdx.x * 8 + wave;
    if (e >= nEdges) return;

    const int s = src[e];
    const int d = dst[e];

    // Look-ahead prefetch of a future edge's source row (global_prefetch_b8).
    const int e2 = e + 4096;
    if (e2 < nEdges) {
        __builtin_prefetch(x + (size_t)src[e2] * HD + lane * 4, 0, 1);
    }

    const float4 v = ((const float4*)(x + (size_t)s * HD))[lane];
    float* ar = agg + (size_t)d * HD + (size_t)lane * 4;
    unsafeAtomicAdd(ar + 0, v.x);
    unsafeAtomicAdd(ar + 1, v.y);
    unsafeAtomicAdd(ar + 2, v.z);
    unsafeAtomicAdd(ar + 3, v.w);
}

// ---------------------------------------------------------------------------
// WMMA fp32 GEMM, compile-time shapes, fused bias + BatchNorm + ReLU.
//   Aeff = HAS_A2 ? (1+*epsPtr)*A + A2 : A                [M x KDIM]
//   C    = relu( BN( Aeff @ Bw + bias ) )                 [M x NDIM]
// Block = 256 threads (8 waves). Grid: (M/16, NDIM/128).
// A tile (16 x KDIM) staged in LDS once per block, shared by all 8 waves;
// each wave computes one 16x16 C tile at tn = blockIdx.y*128 + wave*16.
// No divergent control flow -> EXEC all ones for WMMA.
// ---------------------------------------------------------------------------
template<int NDIM, int KDIM, bool HAS_A2>
__global__ __launch_bounds__(256)
void wmma_gemm_bn_relu(const float* __restrict__ A,
                       const float* __restrict__ A2,
                       const float* __restrict__ epsPtr,
                       const float* __restrict__ Bw,
                       const float* __restrict__ bias,
                       const float* __restrict__ g,
                       const float* __restrict__ be,
                       const float* __restrict__ rm,
                       const float* __restrict__ rv,
                       float* __restrict__ C)
{
    constexpr int LDK = KDIM + 4;               // pad: bank spread, 16B align
    __shared__ float As[16 * LDK];

    const int lane = threadIdx.x & 31;
    const int wave = threadIdx.x >> 5;
    const int half = lane >> 4;                 // 0: lanes 0-15, 1: 16-31
    const int l16  = lane & 15;

    const int tm = blockIdx.x * 16;
    const int tn = blockIdx.y * 128 + wave * 16;
    const int col = tn + l16;

    // ---- stage A tile (16 x KDIM) into LDS ----
    constexpr int NV4 = (16 * KDIM) / 4;        // float4 elements in tile
    if (HAS_A2) {
        const float coef = 1.0f + epsPtr[0];
#pragma unroll
        for (int i = threadIdx.x; i < NV4; i += 256) {
            const int r = i / (KDIM / 4);
            const int c = (i % (KDIM / 4)) * 4;
            const size_t go = (size_t)(tm + r) * KDIM + c;
            const float4 xa = *(const float4*)(A + go);
            const float4 ag = *(const float4*)(A2 + go);
            float4 v;
            v.x = coef * xa.x + ag.x;
            v.y = coef * xa.y + ag.y;
            v.z = coef * xa.z + ag.z;
            v.w = coef * xa.w + ag.w;
            *(float4*)&As[r * LDK + c] = v;
        }
    } else {
#pragma unroll
        for (int i = threadIdx.x; i < NV4; i += 256) {
            const int r = i / (KDIM / 4);
            const int c = (i % (KDIM / 4)) * 4;
            copy16_g2l(A + (size_t)(tm + r) * KDIM + c, &As[r * LDK + c]);
        }
        wait_async_lds();
    }
    __syncthreads();

    // ---- K loop: fully unrolled WMMA chain ----
    const float* aLds = &As[l16 * LDK];         // this lane's A row
    const float* bCol = Bw + col;               // this lane's B column
    v8f acc = {0.f, 0.f, 0.f, 0.f, 0.f, 0.f, 0.f, 0.f};

#pragma unroll
    for (int k = 0; k < KDIM; k += 4) {
        const int kk = k + 2 * half;            // lanes0-15: K,K+1; 16-31: K+2,K+3
        const v2f a = *(const v2f*)(aLds + kk); // ds_load_b64 (8B aligned)
        v2f b;
        b.x = bCol[(size_t)kk * NDIM];
        b.y = bCol[(size_t)(kk + 1) * NDIM];
        acc = __builtin_amdgcn_wmma_f32_16x16x4_f32(
                  false, a, false, b, (short)0, acc, false, false);
    }

    // ---- fused epilogue: (acc + bias) -> BN -> ReLU ----
    const float bi  = bias[col];
    const float inv = rsqrtf(rv[col] + BN_EPS);
    const float scale = g[col] * inv;
    const float shift = be[col] - rm[col] * scale;
#pragma unroll
    for (int r = 0; r < 8; ++r) {
        float v = (acc[r] + bi) * scale + shift;
        v = fmaxf(v, 0.0f);
        // C/D layout: VGPR r -> row r (lanes 0-15) / row r+8 (lanes 16-31)
        C[(size_t)(tm + r + 8 * half) * NDIM + col] = v;
    }
}

// ---------------------------------------------------------------------------
// SAGPool score: s = tanh( agg·Wrel + brel + x·Wroot ). One wave per node.
// ---------------------------------------------------------------------------
__global__ __launch_bounds__(256)
void sag_score(const float* __restrict__ x,
               const float* __restrict__ agg,
               const float* __restrict__ Wrel,
               const float* __restrict__ brel,
               const float* __restrict__ Wroot,
               float* __restrict__ scores)
{
    const int lane = threadIdx.x & 31;
    const int wave = threadIdx.x >> 5;
    const int node = blockIdx.x * 8 + wave;

    const float4 xv = ((const float4*)(x   + (size_t)node * HD))[lane];
    const float4 av = ((const float4*)(agg + (size_t)node * HD))[lane];
    const float4 wv = ((const float4*)Wrel)[lane];
    const float4 ov = ((const float4*)Wroot)[lane];

    float p = av.x * wv.x + av.y * wv.y + av.z * wv.z + av.w * wv.w
            + xv.x * ov.x + xv.y * ov.y + xv.z * ov.z + xv.w * ov.w;

    for (int off = 16; off; off >>= 1) p += __shfl_xor(p, off, 32);
    if (lane == 0) scores[node] = tanhf(p + brel[0]);
}

// ---------------------------------------------------------------------------
// Per-graph top-K selection (exact rank, lax.top_k tie ordering) + weighted
// global-add-pool. One 512-thread block per graph. Output order-invariant.
// ---------------------------------------------------------------------------
__global__ __launch_bounds__(512)
void topk_pool(const float* __restrict__ x,
               const float* __restrict__ scores,
               float* __restrict__ emb)
{
    __shared__ float s[NNODES];
    __shared__ int   ids[KPOOL];
    __shared__ float vals[KPOOL];
    __shared__ int   counter;
    __shared__ float part[4][HD];

    const int b = blockIdx.x;
    const int t = threadIdx.x;
    if (t == 0) counter = 0;
    s[t] = scores[b * NNODES + t];
    __syncthreads();

    const float si = s[t];
    int rank = 0;
    for (int j = 0; j < NNODES; ++j) {
        const float sj = s[j];
        rank += (sj > si) || (sj == si && j < t);   // lower index wins ties
    }
    if (rank < KPOOL) {
        const int slot = atomicAdd(&counter, 1);
        ids[slot]  = b * NNODES + t;                // global node id
        vals[slot] = si;
    }
    __syncthreads();

    // emb[b][c] = sum_{k selected} x[id_k][c] * val_k  (4-way split, no atomics)
    const int c = t & (HD - 1);
    const int grp = t >> 7;                         // 0..3
    float acc = 0.0f;
    for (int k = grp; k < KPOOL; k += 4)
        acc += x[(size_t)ids[k] * HD + c] * vals[k];
    part[grp][c] = acc;
    __syncthreads();
    if (grp == 0)
        emb[b * HD + c] = part[0][c] + part[1][c] + part[2][c] + part[3][c];
}

// ---------------------------------------------------------------------------
// Final tiny projection: logits = h @ W_lin2 + b_lin2   (128 x 10)
// ---------------------------------------------------------------------------
__global__ void lin2_kernel(const float* __restrict__ h,
                            const float* __restrict__ W,
                            const float* __restrict__ bias,
                            float* __restrict__ out)
{
    const int i = blockIdx.x * blockDim.x + threadIdx.x;   // b*NOUT + o
    if (i >= NGRAPH * NOUT) return;
    const int b = i / NOUT, o = i % NOUT;
    float acc = bias[o];
    for (int k = 0; k < HD; ++k)
        acc += h[b * HD + k] * W[k * NOUT + o];
    out[i] = acc;
}

// ---------------------------------------------------------------------------
extern "C" void kernel_launch(void* const* d_in, const int* in_sizes, int n_in,
                              void* d_out, int out_size, void* d_ws, size_t ws_size,
                              hipStream_t stream)
{
    const float* x0   = (const float*)d_in[0];
    const int*   ei   = (const int*)d_in[1];
    // d_in[2] = batch (unused: pooling is done per-graph directly)
    const float* epss = (const float*)d_in[3];
    const float* Ws1  = (const float*)d_in[4];
    const float* bs1  = (const float*)d_in[5];
    const float* g1s  = (const float*)d_in[6];
    const float* be1s = (const float*)d_in[7];
    const float* rm1s = (const float*)d_in[8];
    const float* rv1s = (const float*)d_in[9];
    const float* Ws2  = (const float*)d_in[10];
    const float* bs2  = (const float*)d_in[11];
    const float* gbn  = (const float*)d_in[12];
    const float* bbn  = (const float*)d_in[13];
    const float* rmbn = (const float*)d_in[14];
    const float* rvbn = (const float*)d_in[15];
    const float* Wrel = (const float*)d_in[16];
    const float* brel = (const float*)d_in[17];
    const float* Wroot= (const float*)d_in[18];
    const float* Wl1  = (const float*)d_in[19];
    const float* bl1  = (const float*)d_in[20];
    const float* g_h  = (const float*)d_in[21];
    const float* be_h = (const float*)d_in[22];
    const float* rm_h = (const float*)d_in[23];
    const float* rv_h = (const float*)d_in[24];
    const float* Wl2  = (const float*)d_in[25];
    const float* bl2  = (const float*)d_in[26];

    const int nEdges = in_sizes[1] / 2;
    const int* src = ei;
    const int* dst = ei + nEdges;

    // Workspace carve (~168 MB): agg | h1 | x1 | x2 | scores | hhead
    float* agg    = (float*)d_ws;                         // NT*128
    float* h1     = agg    + (size_t)NT_TOTAL * HD;       // NT*256
    float* x1     = h1     + (size_t)NT_TOTAL * H2;       // NT*128
    float* x2     = x1     + (size_t)NT_TOTAL * HD;       // NT*128
    float* scores = x2     + (size_t)NT_TOTAL * HD;       // NT
    float* hhead  = scores + NT_TOTAL;                    // 128*128

    float* emb    = (float*)d_out;                        // [128,128]
    float* logits = emb + NGRAPH * HD;                    // [128,10]

    const dim3 blk256(256);
    const int scatterBlocks = (nEdges + 7) / 8;
    const size_t aggBytes = (size_t)NT_TOTAL * HD * sizeof(float);

    const float* xin = x0;
    float* xouts[2] = {x1, x2};

    for (int l = 0; l < 2; ++l) {
        (void)hipMemsetAsync(agg, 0, aggBytes, stream);
        gin_scatter_add<<<scatterBlocks, blk256, 0, stream>>>(xin, src, dst, agg, nEdges);

        // h1 = relu(BN1( ((1+eps)x + agg) @ W1 + b1 ))        [NT x 256]
        wmma_gemm_bn_relu<H2, HD, true>
            <<<dim3(NT_TOTAL / 16, H2 / 128), blk256, 0, stream>>>(
            xin, agg, epss + l, Ws1 + (size_t)l * HD * H2, bs1 + l * H2,
            g1s + l * H2, be1s + l * H2, rm1s + l * H2, rv1s + l * H2, h1);

        // x_{l+1} = relu(BN2( h1 @ W2 + b2 ))                  [NT x 128]
        wmma_gemm_bn_relu<HD, H2, false>
            <<<dim3(NT_TOTAL / 16, 1), blk256, 0, stream>>>(
            h1, nullptr, nullptr, Ws2 + (size_t)l * H2 * HD, bs2 + l * HD,
            gbn + l * HD, bbn + l * HD, rmbn + l * HD, rvbn + l * HD,
            xouts[l]);

        xin = xouts[l];
    }

    // ---- SAGPool: GraphConv score -> tanh -> per-graph top-K -> add-pool ----
    (void)hipMemsetAsync(agg, 0, aggBytes, stream);
    gin_scatter_add<<<scatterBlocks, blk256, 0, stream>>>(x2, src, dst, agg, nEdges);
    sag_score<<<NT_TOTAL / 8, blk256, 0, stream>>>(x2, agg, Wrel, brel, Wroot, scores);
    topk_pool<<<NGRAPH, 512, 0, stream>>>(x2, scores, emb);

    // ---- head: relu(BN(emb @ Wl1 + bl1)) @ Wl2 + bl2 ----
    wmma_gemm_bn_relu<HD, HD, false>
        <<<dim3(NGRAPH / 16, 1), blk256, 0, stream>>>(
        emb, nullptr, nullptr, Wl1, bl1, g_h, be_h, rm_h, rv_h, hhead);
    lin2_kernel<<<(NGRAPH * NOUT + 127) / 128, 128, 0, stream>>>(hhead, Wl2, bl2, logits);
}